// GIN_Module_79001628442825
// MI455X (gfx1250) — compile-verified
//
#include <hip/hip_runtime.h>

#define NN 50000
#define NE 800000
#define D  64
#define LDP 65   // padded LDS row stride (floats) to avoid bank conflicts

typedef __attribute__((ext_vector_type(2))) float v2f;
typedef __attribute__((ext_vector_type(8))) float v8f;

// ---------------- h = src (vectorized copy) ----------------
__global__ void gin_copy_kernel(const float* __restrict__ s, float* __restrict__ d, int n4) {
    int i = blockIdx.x * blockDim.x + threadIdx.x;
    if (i < n4) ((float4*)d)[i] = ((const float4*)s)[i];
}

// ---------------- h[dst] += x[src] over edges (segment sum) ----------------
// 16 threads per edge, float4 gather + 4x global_atomic_add_f32 scatter.
__global__ void gin_scatter_kernel(const float* __restrict__ x, const int* __restrict__ src,
                                   const int* __restrict__ dst, float* __restrict__ h) {
    int idx = blockIdx.x * blockDim.x + threadIdx.x;
    int e = idx >> 4;
    if (e >= NE) return;
    int f = (idx & 15) << 2;
    int s = src[e];
    int d = dst[e];
    float4 v = *(const float4*)(x + s * D + f);
    float* p = h + d * D + f;
    unsafeAtomicAdd(p + 0, v.x);
    unsafeAtomicAdd(p + 1, v.y);
    unsafeAtomicAdd(p + 2, v.z);
    unsafeAtomicAdd(p + 3, v.w);
}

// ---------------- out = tanh(h@w1 + b1) @ w2 + b2, fused, f32 WMMA ----------------
// 128 threads = 4 waves; each wave computes a 16x64 row block via 4 N-tiles of
// V_WMMA_F32_16X16X4_F32, K-loop of 16 steps. h tile + weights staged in LDS.
__global__ __launch_bounds__(128, 1) void gin_mlp_kernel(
    const float* __restrict__ h, const float* __restrict__ w1, const float* __restrict__ b1,
    const float* __restrict__ w2, const float* __restrict__ b2, float* __restrict__ out) {

    __shared__ float w1s[D * LDP];
    __shared__ float w2s[D * LDP];
    __shared__ float hts[64 * LDP];   // h tile, then overwritten (wave-private rows) with t = tanh(...)

    const int tid  = threadIdx.x;
    const int wave = tid >> 5;
    const int lane = tid & 31;
    const int n    = lane & 15;   // N index within 16-wide tile; also M index of A fragment
    const int hi   = lane >> 4;   // half-wave select
    const int r0   = blockIdx.x * 64;

    // cooperative staging: weights + 64-row h tile (rows clamped for tail block)
    for (int i = tid; i < D * D; i += 128) {
        int k = i >> 6, c = i & 63;
        w1s[k * LDP + c] = w1[i];
        w2s[k * LDP + c] = w2[i];
        int row = r0 + k;
        if (row >= NN) row = NN - 1;
        hts[k * LDP + c] = h[row * D + c];
    }
    __syncthreads();

    const int tr = wave * 16;     // wave's first row inside the tile

    // ---- GEMM1: acc = h_tile @ w1 ----
    v8f acc[4];
#pragma unroll
    for (int t = 0; t < 4; ++t) acc[t] = (v8f){0.f,0.f,0.f,0.f,0.f,0.f,0.f,0.f};

#pragma unroll
    for (int k0 = 0; k0 < D; k0 += 4) {
        const int ka = k0 + 2 * hi;
        v2f a;                                    // A[m=n][ka], A[m=n][ka+1]
        a.x = hts[(tr + n) * LDP + ka];
        a.y = hts[(tr + n) * LDP + ka + 1];
#pragma unroll
        for (int t = 0; t < 4; ++t) {
            v2f b;                                // B[ka][n+16t], B[ka+1][n+16t]
            b.x = w1s[ka * LDP + t * 16 + n];
            b.y = w1s[(ka + 1) * LDP + t * 16 + n];
            acc[t] = __builtin_amdgcn_wmma_f32_16x16x4_f32(
                false, a, false, b, (short)0, acc[t], false, false);
        }
    }

    // bias + tanh, write t back into this wave's private rows of hts
#pragma unroll
    for (int t = 0; t < 4; ++t) {
        float bv = b1[t * 16 + n];
#pragma unroll
        for (int v = 0; v < 8; ++v) {             // C layout: m = v + 8*hi, col = n + 16*t
            float val = tanhf(acc[t][v] + bv);
            hts[(tr + v + 8 * hi) * LDP + t * 16 + n] = val;
        }
    }
    // no barrier: each wave only reads/writes its own 16 rows of hts

    // ---- GEMM2: acc2 = t_tile @ w2 ----
    v8f acc2[4];
#pragma unroll
    for (int t = 0; t < 4; ++t) acc2[t] = (v8f){0.f,0.f,0.f,0.f,0.f,0.f,0.f,0.f};

#pragma unroll
    for (int k0 = 0; k0 < D; k0 += 4) {
        const int ka = k0 + 2 * hi;
        v2f a;
        a.x = hts[(tr + n) * LDP + ka];
        a.y = hts[(tr + n) * LDP + ka + 1];
#pragma unroll
        for (int t = 0; t < 4; ++t) {
            v2f b;
            b.x = w2s[ka * LDP + t * 16 + n];
            b.y = w2s[(ka + 1) * LDP + t * 16 + n];
            acc2[t] = __builtin_amdgcn_wmma_f32_16x16x4_f32(
                false, a, false, b, (short)0, acc2[t], false, false);
        }
    }

    // bias + guarded store
#pragma unroll
    for (int t = 0; t < 4; ++t) {
        float bv = b2[t * 16 + n];
#pragma unroll
        for (int v = 0; v < 8; ++v) {
            int row = r0 + tr + v + 8 * hi;
            if (row < NN) out[row * D + t * 16 + n] = acc2[t][v] + bv;
        }
    }
}

extern "C" void kernel_launch(void* const* d_in, const int* in_sizes, int n_in,
                              void* d_out, int out_size, void* d_ws, size_t ws_size,
                              hipStream_t stream) {
    const float* x    = (const float*)d_in[0];
    const int*   src  = (const int*)  d_in[1];
    const int*   dst  = (const int*)  d_in[2];
    const float* w1_0 = (const float*)d_in[3];
    const float* b1_0 = (const float*)d_in[4];
    const float* w2_0 = (const float*)d_in[5];
    const float* b2_0 = (const float*)d_in[6];
    const float* w1_1 = (const float*)d_in[7];
    const float* b1_1 = (const float*)d_in[8];
    const float* w2_1 = (const float*)d_in[9];
    const float* b2_1 = (const float*)d_in[10];
    float* out  = (float*)d_out;
    float* hbuf = (float*)d_ws;          // 50000*64*4 = 12.8 MB scratch

    const int n4 = NN * D / 4;
    dim3 cb(256), cg((n4 + 255) / 256);
    dim3 sb(256), sg((NE * 16 + 255) / 256);
    dim3 gb(128), gg((NN + 63) / 64);

    // layer 0: h = x + segsum(x[src] -> dst); out = MLP0(h)
    gin_copy_kernel   <<<cg, cb, 0, stream>>>(x, hbuf, n4);
    gin_scatter_kernel<<<sg, sb, 0, stream>>>(x, src, dst, hbuf);
    gin_mlp_kernel    <<<gg, gb, 0, stream>>>(hbuf, w1_0, b1_0, w2_0, b2_0, out);

    // layer 1: h = y + segsum(y[src] -> dst); out = MLP1(h)
    gin_copy_kernel   <<<cg, cb, 0, stream>>>(out, hbuf, n4);
    gin_scatter_kernel<<<sg, sb, 0, stream>>>(out, src, dst, hbuf);
    gin_mlp_kernel    <<<gg, gb, 0, stream>>>(hbuf, w1_1, b1_1, w2_1, b2_1, out);
}